// YOLOLoss_13924283973748
// MI455X (gfx1250) — compile-verified
//
#include <hip/hip_runtime.h>
#include <math.h>

#define NCLS 80
#define NGTS 32
#define NANC 3
#define BATCH 16

typedef __attribute__((ext_vector_type(2))) float v2f;
typedef __attribute__((ext_vector_type(8))) float v8f;

__device__ __forceinline__ float softplusf(float x) {
  // numerically stable: max(x,0) + log1p(exp(-|x|))
  return fmaxf(x, 0.0f) + log1pf(expf(-fabsf(x)));
}

// Workspace accumulator layout, per level L (8 floats):
//   [L*8+0] npos, [L*8+1] box_sq_sum, [L*8+2] obj_p_at_pos_sum,
//   [L*8+3] cls_bce_sum, [L*8+4] softplus(obj_p) sum over all cells
__global__ void init_accums_kernel(float* acc) {
  int i = threadIdx.x;
  if (i < 24) acc[i] = 0.0f;
}

// One block (1 wave, 32 threads) per image. Does GT->cell assignment and
// accumulates the positive-cell loss terms directly.
__global__ void assign_pos_kernel(const float* __restrict__ pred,
                                  const float* __restrict__ anc,
                                  const float* __restrict__ gt_boxes,
                                  const int* __restrict__ gt_labels,
                                  float* __restrict__ acc,
                                  int H, int W, float stride, int level) {
  __shared__ int   skey[NGTS];
  __shared__ float siou[NGTS];
  const int b = blockIdx.x;
  const int n = threadIdx.x;          // GT index 0..31
  const int HW = H * W;

  const float* gb = gt_boxes + ((size_t)b * NGTS + n) * 4;
  const float x1 = gb[0], y1 = gb[1], x2 = gb[2], y2 = gb[3];
  const float cx = 0.5f * (x1 + x2), cy = 0.5f * (y1 + y2);
  int gx = (int)(cx / stride); gx = min(max(gx, 0), W - 1);
  int gy = (int)(cy / stride); gy = min(max(gy, 0), H - 1);
  const float area1 = (x2 - x1) * (y2 - y1);

  float best_iou = -1.0f; int best_a = 0;
  for (int a = 0; a < NANC; ++a) {                 // argmax, first-max on tie
    const float* ac = anc + (((size_t)a * H + gy) * W + gx) * 4;
    const float acx = ac[0], acy = ac[1], aw = ac[2], ah = ac[3];
    const float lx = fmaxf(x1, acx - 0.5f * aw);
    const float ly = fmaxf(y1, acy - 0.5f * ah);
    const float rx = fminf(x2, acx + 0.5f * aw);
    const float ry = fminf(y2, acy + 0.5f * ah);
    const float inter = fmaxf(rx - lx, 0.0f) * fmaxf(ry - ly, 0.0f);
    const float iou = inter / (area1 + aw * ah - inter + 1e-16f);
    if (iou > best_iou) { best_iou = iou; best_a = a; }
  }
  const int key = best_a * HW + gy * W + gx;
  skey[n] = key; siou[n] = best_iou;
  __syncthreads();

  // winner: iou >= max iou among GTs sharing this key; chosen: min index winner
  float maxiou = best_iou;
  for (int j = 0; j < NGTS; ++j)
    if (skey[j] == key) maxiou = fmaxf(maxiou, siou[j]);
  bool chosen = (best_iou >= maxiou);
  if (chosen) {
    for (int j = 0; j < n; ++j)
      if (skey[j] == key && siou[j] >= maxiou) { chosen = false; break; }
  }

  if (chosen) {
    const float* ak = anc + (size_t)key * 4;       // anchors.reshape(K,4)[key]
    const float gw = x2 - x1, gh = y2 - y1;
    const float tx = (cx - ak[0]) * 0.125f;        // s0 = 8 for ALL levels (ref)
    const float ty = (cy - ak[1]) * 0.125f;
    const float tw = logf(gw / ak[2] + 1e-16f);
    const float th = logf(gh / ak[3] + 1e-16f);

    const size_t base = ((size_t)b * (NANC * HW) + key) * (5 + NCLS);
    const float d0 = pred[base + 0] - tx, d1 = pred[base + 1] - ty;
    const float d2 = pred[base + 2] - tw, d3 = pred[base + 3] - th;
    const float bsum = d0 * d0 + d1 * d1 + d2 * d2 + d3 * d3;
    const float objp = pred[base + 4];

    const int lab = gt_labels[b * NGTS + n];
    float csum = 0.0f;
    for (int c = 0; c < NCLS; ++c) {
      const float z = pred[base + 5 + c];
      csum += softplusf(z) - ((c == lab) ? z : 0.0f);
    }
    float* a = acc + level * 8;
    atomicAdd(&a[0], 1.0f);
    atomicAdd(&a[1], bsum);
    atomicAdd(&a[2], objp);
    atomicAdd(&a[3], csum);
  }
}

// Bulk work: sum of softplus(obj logits) over every cell of every level.
// Wave reduction done with one V_WMMA_F32_16X16X4_F32 (B = ones).
__global__ void obj_softplus_kernel(const float* __restrict__ p0,
                                    const float* __restrict__ p1,
                                    const float* __restrict__ p2,
                                    float* __restrict__ acc) {
  const int lvl = blockIdx.y;
  const float* p = (lvl == 0) ? p0 : ((lvl == 1) ? p1 : p2);
  const int n = (lvl == 0) ? 307200 : ((lvl == 1) ? 76800 : 19200);

  float s = 0.0f;
  const int gstride = gridDim.x * blockDim.x;
  for (int i = blockIdx.x * blockDim.x + threadIdx.x; i < n; i += gstride) {
    const size_t idx = (size_t)i * 85 + 4;
    if (i + gstride < n)
      __builtin_prefetch(&p[idx + (size_t)gstride * 85], 0, 0);
    s += softplusf(p[idx]);
  }

  // ---- wave32 sum via WMMA f32 16x16x4: A[m][0]=s_m, A[m][2]=s_{m+16}, B=1 ----
  float t;
#if __has_builtin(__builtin_amdgcn_wmma_f32_16x16x4_f32)
  v2f av; av[0] = s;    av[1] = 0.0f;
  v2f bv; bv[0] = 1.0f; bv[1] = 1.0f;
  v8f cv = {};
  cv = __builtin_amdgcn_wmma_f32_16x16x4_f32(false, av, false, bv,
                                             (short)0, cv, false, false);
  // lanes 0-15 hold rowsum[0..7] in their 8 VGPRs, lanes 16-31 hold rowsum[8..15]
  t = cv[0] + cv[1] + cv[2] + cv[3] + cv[4] + cv[5] + cv[6] + cv[7];
  t += __shfl_xor(t, 16, 32);    // combine the two halves -> full 32-lane sum
#else
  t = s;
  for (int off = 16; off > 0; off >>= 1) t += __shfl_xor(t, off, 32);
#endif

  __shared__ float wsum[8];
  const int wid = threadIdx.x >> 5;
  if ((threadIdx.x & 31) == 0) wsum[wid] = t;
  __syncthreads();
  if (threadIdx.x == 0) {
    float bs = 0.0f;
    const int nw = blockDim.x >> 5;
    for (int w = 0; w < nw; ++w) bs += wsum[w];
    atomicAdd(&acc[lvl * 8 + 4], bs);
  }
}

__global__ void finalize_kernel(const float* __restrict__ acc, float* __restrict__ out) {
  if (threadIdx.x != 0) return;
  const float cells[3] = {307200.0f, 76800.0f, 19200.0f};
  float tb = 0.0f, tc = 0.0f, to = 0.0f, tot = 0.0f;
  for (int L = 0; L < 3; ++L) {
    const float* a = acc + L * 8;
    const float npos  = a[0];
    const float denom = fmaxf(npos, 1.0f);
    const float box_l = a[1] / denom;
    const float cls_l = a[3] / (denom * (float)NCLS);
    const float obj_l = (a[4] - a[2]) / cells[L];   // mean BCE = (Σsp - Σ_pos x)/N
    if (npos > 0.0f) { tb += box_l; tc += cls_l; }
    to += obj_l;
    tot += npos;
  }
  const float fb = (tot > 0.0f) ? tb / 3.0f : 0.0f;
  const float fc = (tot > 0.0f) ? tc / 3.0f : 0.0f;
  const float fo = to / 3.0f;
  out[0] = 0.05f * fb + 1.0f * fo + 0.5f * fc;
}

extern "C" void kernel_launch(void* const* d_in, const int* in_sizes, int n_in,
                              void* d_out, int out_size, void* d_ws, size_t ws_size,
                              hipStream_t stream) {
  // setup_inputs() dict order: p0, a0, p1, a1, p2, a2, gt_boxes, gt_labels
  const float* p0  = (const float*)d_in[0];
  const float* a0  = (const float*)d_in[1];
  const float* p1  = (const float*)d_in[2];
  const float* a1  = (const float*)d_in[3];
  const float* p2  = (const float*)d_in[4];
  const float* a2  = (const float*)d_in[5];
  const float* gtb = (const float*)d_in[6];
  const int*   gtl = (const int*)d_in[7];
  float* acc = (float*)d_ws;
  float* out = (float*)d_out;

  init_accums_kernel<<<1, 32, 0, stream>>>(acc);
  assign_pos_kernel<<<BATCH, 32, 0, stream>>>(p0, a0, gtb, gtl, acc, 80, 80,  8.0f, 0);
  assign_pos_kernel<<<BATCH, 32, 0, stream>>>(p1, a1, gtb, gtl, acc, 40, 40, 16.0f, 1);
  assign_pos_kernel<<<BATCH, 32, 0, stream>>>(p2, a2, gtb, gtl, acc, 20, 20, 32.0f, 2);
  dim3 og(256, 3);
  obj_softplus_kernel<<<og, 256, 0, stream>>>(p0, p1, p2, acc);
  finalize_kernel<<<1, 32, 0, stream>>>(acc, out);
}